// MoE_62139586839405
// MI455X (gfx1250) — compile-verified
//
#include <hip/hip_runtime.h>
#include <math.h>

typedef float v2f __attribute__((ext_vector_type(2)));
typedef float v4f __attribute__((ext_vector_type(4)));
typedef float v8f __attribute__((ext_vector_type(8)));
typedef int   v4i __attribute__((vector_size(16)));   // matches builtin's int __vector(4)

#define B_SZ   64
#define E_SZ   8
#define DIN    320
#define NP     196      // patches per image (14x14)
#define DMODEL 768      // out features
#define KDIM   768      // in features = 3*16*16
#define PATCH  16
#define IMG    224
#define GW     14       // IMG/PATCH

#define KC     16       // K chunk staged in LDS
#define APAD   20       // padded LDS row stride (words); 20*k mod 64 -> conflict-free
#define NT     128      // N tile per block
#define MTP    208      // M padded to 13*16
#define NCHUNK (KDIM / KC)   // 48

// LDS layout (floats): A buf0 | A buf1 | B buf0 | B buf1 | dump
#define AOFF0  0
#define AOFF1  (MTP * APAD)                  // 4160
#define BOFF0  (2 * MTP * APAD)              // 8320
#define BOFF1  (2 * MTP * APAD + NT * APAD)  // 10880
#define DUMPO  (2 * MTP * APAD + 2 * NT * APAD) // 13440
#define LDS_WORDS (DUMPO + 32 * 4)           // 13568 -> 54272 B

#define EPS_COMBINE 2.220446049250313e-16f

// ---- CDNA5 async global->LDS copy (ASYNCcnt) with portable fallback ----
#if defined(__has_builtin)
#  if __has_builtin(__builtin_amdgcn_global_load_async_to_lds_b128)
#    define HAVE_ASYNC_LDS 1
#  endif
#endif
#ifndef HAVE_ASYNC_LDS
#  define HAVE_ASYNC_LDS 0
#endif

#if HAVE_ASYNC_LDS
#  define ASYNC_CP16(g, l)                                                     \
     __builtin_amdgcn_global_load_async_to_lds_b128(                           \
         (__attribute__((address_space(1))) v4i*)(g),                          \
         (__attribute__((address_space(3))) v4i*)(l), 0, 0)
#  if __has_builtin(__builtin_amdgcn_s_wait_asynccnt)
#    define WAIT_ASYNC(n) __builtin_amdgcn_s_wait_asynccnt(n)
#  else
#    define WAIT_ASYNC(n) asm volatile("s_wait_asynccnt %0" ::"n"(n) : "memory")
#  endif
#else
#  define ASYNC_CP16(g, l) (*(v4f*)(l) = *(const v4f*)(g))
#  define WAIT_ASYNC(n) do {} while (0)
#endif

// ---------------- gating: noisy top-1, load-balance loss ----------------
__global__ void gating_kernel(const float* __restrict__ g,
                              const float* __restrict__ noise,
                              const float* __restrict__ w_gate,
                              const float* __restrict__ w_noise,
                              int*   __restrict__ sel,
                              float* __restrict__ loss_out)
{
    __shared__ float probs[B_SZ][E_SZ];
    __shared__ int   sels[B_SZ];

    int b = threadIdx.x;
    if (b < B_SZ) {
        float clean[E_SZ], nstd[E_SZ], noisy[E_SZ];
        for (int e = 0; e < E_SZ; ++e) { clean[e] = 0.f; nstd[e] = 0.f; }
        for (int i = 0; i < DIN; ++i) {
            float gv = g[b * DIN + i];
            for (int e = 0; e < E_SZ; ++e) {
                clean[e] += gv * w_gate [i * E_SZ + e];
                nstd [e] += gv * w_noise[i * E_SZ + e];
            }
        }
        for (int e = 0; e < E_SZ; ++e) {
            float s = nstd[e];
            s = (s > 20.f) ? s : log1pf(expf(s));   // softplus
            s += 0.01f;                              // NOISE_EPS
            nstd[e]  = s;
            noisy[e] = clean[e] + noise[b * E_SZ + e] * s;
        }
        int   i1 = 0; float v1 = noisy[0];
        for (int e = 1; e < E_SZ; ++e) if (noisy[e] > v1) { v1 = noisy[e]; i1 = e; }
        float v2 = -INFINITY;
        for (int e = 0; e < E_SZ; ++e) if (e != i1 && noisy[e] > v2) v2 = noisy[e];

        sels[b] = i1;
        sel[b]  = i1;

        const float inv_sqrt2 = 0.70710678118654752440f;
        for (int e = 0; e < E_SZ; ++e) {
            float thr = (noisy[e] > v2) ? v2 : v1;   // thr_in : thr_out
            float z   = (clean[e] - thr) / nstd[e];
            probs[b][e] = 0.5f * erfcf(-z * inv_sqrt2);   // Phi(z)
        }
    }
    __syncthreads();

    if (threadIdx.x == 0) {
        float load[E_SZ], imp[E_SZ];
        for (int e = 0; e < E_SZ; ++e) { load[e] = 0.f; imp[e] = 0.f; }
        for (int bb = 0; bb < B_SZ; ++bb) {          // deterministic order
            for (int e = 0; e < E_SZ; ++e) load[e] += probs[bb][e];
            imp[sels[bb]] += 1.0f;
        }
        float cv[2];
        const float* arr[2] = { imp, load };
        for (int a = 0; a < 2; ++a) {
            float m = 0.f;
            for (int e = 0; e < E_SZ; ++e) m += arr[a][e];
            m /= (float)E_SZ;
            float var = 0.f;
            for (int e = 0; e < E_SZ; ++e) { float d = arr[a][e] - m; var += d * d; }
            var /= (float)(E_SZ - 1);                // ddof=1
            cv[a] = var / (m * m + 1e-10f);
        }
        *loss_out = (cv[0] + cv[1]) * 1e-2f;
    }
}

// ------- fused patchify + selected-expert GEMM (FP32 WMMA, async double-buffer) -------
__global__ __launch_bounds__(256) void moe_gemm_kernel(
    const float* __restrict__ x,
    const float* __restrict__ w_exp,
    const float* __restrict__ b_exp,
    const int*   __restrict__ sel,
    float*       __restrict__ y)
{
    __shared__ float L[LDS_WORDS];

    const int b  = blockIdx.y;
    const int n0 = blockIdx.x * NT;
    const int e  = sel[b];

    const float* xb    = x     + (size_t)b * 3 * IMG * IMG;
    const float* wbase = w_exp + (size_t)e * DMODEL * KDIM;

    const int tid  = threadIdx.x;
    const int wave = tid >> 5;
    const int lane = tid & 31;
    const int l15  = lane & 15;
    const int half = (lane >> 4) << 1;     // 0 for lanes 0-15, 2 for lanes 16-31

    // --- per-thread staging descriptors (constant across chunks) ---
    // A: 832 float4-items (208 rows x 4), wave w owns flat items [w*104, w*104+104)
    const float* pAg[4];
    int          ldsA[4];
    bool         valA[4];
    for (int t = 0; t < 4; ++t) {
        int j  = t * 32 + lane;
        int gi = wave * 104 + j;
        int p  = gi >> 2;
        int i4 = (gi & 3) << 2;            // px offset within patch row (mult of 4)
        bool v = (j < 104) && (p < NP);
        valA[t] = v;
        if (v) {
            int gh = p / GW, gw = p % GW;
            pAg[t]  = xb + (size_t)(gh * PATCH) * IMG + gw * PATCH + i4;
            ldsA[t] = p * APAD + i4;
        } else {
            pAg[t]  = xb;                   // safe dummy global address
            ldsA[t] = DUMPO + lane * 4;     // dump slot (buffer-independent)
        }
    }
    // B: 512 float4-items (128 rows x 4), wave w owns items [w*64, w*64+64)
    const float* pBg[2];
    int          ldsB[2];
    for (int t = 0; t < 2; ++t) {
        int j  = wave * 64 + t * 32 + lane;
        int n  = j >> 2;
        int i4 = (j & 3) << 2;
        pBg[t]  = wbase + (size_t)(n0 + n) * KDIM + i4;
        ldsB[t] = n * APAD + i4;
    }

    // pre-zero A padding rows (196..207) of both buffers; never re-written
    for (int idx = tid; idx < (MTP - NP) * APAD; idx += 256) {
        L[AOFF0 + NP * APAD + idx] = 0.f;
        L[AOFF1 + NP * APAD + idx] = 0.f;
    }

    v8f acc[13];
    for (int mt = 0; mt < 13; ++mt)
        for (int r = 0; r < 8; ++r) acc[mt][r] = 0.f;

    const int aoffs[2] = { AOFF0, AOFF1 };
    const int boffs[2] = { BOFF0, BOFF1 };

    // stage(buf, k0): exactly 6 full-EXEC async b128 per wave
#define STAGE(aoff, boff, k0)                                                  \
    do {                                                                       \
        int _co = ((k0) >> 8) * (IMG * IMG) + (((k0) >> 4) & 15) * IMG;        \
        for (int t = 0; t < 4; ++t)                                            \
            ASYNC_CP16(pAg[t] + _co,                                           \
                       &L[ldsA[t] + (valA[t] ? (aoff) : 0)]);                  \
        for (int t = 0; t < 2; ++t)                                            \
            ASYNC_CP16(pBg[t] + (k0), &L[ldsB[t] + (boff)]);                   \
    } while (0)

    STAGE(aoffs[0], boffs[0], 0);          // prologue: chunk 0 -> buf 0

    for (int k = 0; k < NCHUNK; ++k) {
        const int cur = k & 1;
        if (k + 1 < NCHUNK) {
            STAGE(aoffs[cur ^ 1], boffs[cur ^ 1], (k + 1) * KC);
            WAIT_ASYNC(6);                 // in-order completion: chunk k drained
        } else {
            WAIT_ASYNC(0);
        }
        __syncthreads();                    // all waves' chunk-k data in LDS

        const float* Ab = &L[aoffs[cur]];
        const float* Bb = &L[boffs[cur]];
        for (int kk = 0; kk < KC; kk += 4) {
            v2f bfrag = *(const v2f*)&Bb[(wave * 16 + l15) * APAD + kk + half];
            for (int mt = 0; mt < 13; ++mt) {
                v2f afrag = *(const v2f*)&Ab[(mt * 16 + l15) * APAD + kk + half];
                acc[mt] = __builtin_amdgcn_wmma_f32_16x16x4_f32(
                    /*neg_a=*/false, afrag, /*neg_b=*/false, bfrag,
                    /*c_mod=*/(short)0, acc[mt],
                    /*reuse_a=*/false, /*reuse_b=*/false);
            }
        }
        __syncthreads();                    // buf reusable for chunk k+2
    }
#undef STAGE

    // --- store: C/D layout: VGPR r -> M = r (lanes 0-15) / 8+r (lanes 16-31) ---
    const int   col  = n0 + wave * 16 + l15;
    const float bias = b_exp[e * DMODEL + col];
    const int rbase  = (lane >> 4) * 8;
    for (int mt = 0; mt < 13; ++mt) {
        for (int r = 0; r < 8; ++r) {
            int m = mt * 16 + rbase + r;
            if (m < NP) {
                float v = acc[mt][r] + bias;     // gate == 1.0 exactly (K=1)
                if (v == 0.0f) v = EPS_COMBINE;
                y[((size_t)b * NP + m) * DMODEL + col] = v;
            }
        }
    }
}

extern "C" void kernel_launch(void* const* d_in, const int* in_sizes, int n_in,
                              void* d_out, int out_size, void* d_ws, size_t ws_size,
                              hipStream_t stream) {
    const float* x       = (const float*)d_in[0];
    const float* g       = (const float*)d_in[1];
    const float* noise   = (const float*)d_in[2];
    const float* w_gate  = (const float*)d_in[3];
    const float* w_noise = (const float*)d_in[4];
    const float* w_exp   = (const float*)d_in[5];
    const float* b_exp   = (const float*)d_in[6];

    float* y        = (float*)d_out;
    float* loss_out = y + (size_t)B_SZ * NP * DMODEL;   // loss appended after y
    int*   sel      = (int*)d_ws;                       // 64 ints of scratch

    hipLaunchKernelGGL(gating_kernel, dim3(1), dim3(64), 0, stream,
                       g, noise, w_gate, w_noise, sel, loss_out);
    hipLaunchKernelGGL(moe_gemm_kernel, dim3(DMODEL / NT, B_SZ), dim3(256), 0, stream,
                       x, w_exp, b_exp, sel, y);
}